// GSN_matrix_8392366096424
// MI455X (gfx1250) — compile-verified
//
#include <hip/hip_runtime.h>
#include <math.h>

typedef __attribute__((ext_vector_type(2))) float v2f;
typedef __attribute__((ext_vector_type(8))) float v8f;

static constexpr int kRows = 1025;   // B*N + 1
static constexpr int kD    = 128;    // hidden dim
static constexpr int kTD   = 384;    // 3*D
static constexpr int kB    = 8;
static constexpr int kN    = 128;
static constexpr int kTilesM = 65;   // ceil(1025/16)
static constexpr int kTilesN = 24;   // 384/16

// ---------------------------------------------------------------------------
// G[r][n] = sum_k H[r][k] * W[n][k] + bias[n]    (i.e. G = H @ W^T + b)
// H: [1025 x 128], W: [384 x 128], G: [1025 x 384]
// One wave (32 lanes) computes one 16x16 tile of G using V_WMMA_F32_16X16X4_F32.
// A-matrix (16x4 f32):  lane l -> row M = l%16, K pair (k + 2*(l>>4), +1)  [ISA 7.12.2]
// B-matrix (4x16 f32):  lane l -> col N = l%16, K pair (k + 2*(l>>4), +1)
//                       B[k][n] = W[n][k] => contiguous float2 along W row.
// D (16x16 f32, 8 VGPRs): lane l -> N = l%16, M = v + 8*(l>>4)
// ---------------------------------------------------------------------------
__global__ __launch_bounds__(32)
void gemm_hxt_wmma(const float* __restrict__ H, const float* __restrict__ W,
                   const float* __restrict__ bias, float* __restrict__ G)
{
    const int tile = blockIdx.x;
    const int tm = tile % kTilesM;
    const int tn = tile / kTilesM;
    const int lane = threadIdx.x;
    const int half = lane >> 4;      // 0 or 1
    const int l15  = lane & 15;
    const int m0 = tm * 16;
    const int n0 = tn * 16;

    int arow = m0 + l15;
    if (arow > kRows - 1) arow = kRows - 1;          // clamp ragged M edge
    const float* Ap = H + arow * kD + 2 * half;
    const float* Bp = W + (n0 + l15) * kD + 2 * half;

    v8f acc = {};
    #pragma unroll
    for (int k = 0; k < kD; k += 4) {
        v2f a = *(const v2f*)(Ap + k);
        v2f b = *(const v2f*)(Bp + k);
        acc = __builtin_amdgcn_wmma_f32_16x16x4_f32(
            /*neg_a=*/false, a, /*neg_b=*/false, b,
            /*c_mod=*/(short)0, acc, /*reuse_a=*/false, /*reuse_b=*/false);
    }

    const float bv = bias[n0 + l15];
    #pragma unroll
    for (int v = 0; v < 8; ++v) {
        int m = m0 + v + 8 * half;
        if (m < kRows) G[m * kTD + n0 + l15] = acc[v] + bv;
    }
}

// ---------------------------------------------------------------------------
// One block per output row (b,j) of `change` (plus block 0 = passthrough row).
// change[b,j,:] = sum_i GRU(x = table-row xi, h = table-row hi), with the
// matmul parts pre-factored into Gi/Gh gathers. hout = hin + [0; change].
// input_is_child=1 -> x = child, h = parent (bwd phase); 0 -> swapped (fwd).
// ---------------------------------------------------------------------------
__global__ __launch_bounds__(128)
void gru_gather_phase(const float* __restrict__ hin,
                      const float* __restrict__ Gi,
                      const float* __restrict__ Gh,
                      const int* __restrict__ pidx,
                      const int* __restrict__ cidx,
                      float* __restrict__ hout,
                      int input_is_child)
{
    __shared__ int sp[kN];
    __shared__ int sc[kN];

    const int row = blockIdx.x;
    const int d   = threadIdx.x;

    if (row == 0) {                 // uniform per block: no divergent barrier
        hout[d] = hin[d];
        return;
    }

    const int b = (row - 1) >> 7;
    const int j = (row - 1) & 127;
    const int* pb = pidx + b * kN * kN + j;   // stride kN over i
    const int* cb = cidx + b * kN * kN + j;
    sp[d] = pb[d * kN];
    sc[d] = cb[d * kN];
    __syncthreads();

    float acc = 0.f;
    for (int i = 0; i < kN; ++i) {
        const int pi = sp[i];
        const int ci = sc[i];
        const int xi = input_is_child ? ci : pi;   // row feeding gi (input x)
        const int hi = input_is_child ? pi : ci;   // row feeding gh and h
        const float* gi = Gi + xi * kTD;
        const float* gh = Gh + hi * kTD;
        float ir = gi[d],        hr = gh[d];
        float iz = gi[kD + d],   hz = gh[kD + d];
        float inn = gi[2*kD + d], hn = gh[2*kD + d];
        float h  = hin[hi * kD + d];
        float r = 1.f / (1.f + __expf(-(ir + hr)));
        float z = 1.f / (1.f + __expf(-(iz + hz)));
        float n = tanhf(inn + r * hn);
        acc += (1.f - z) * n + z * h;
    }
    hout[row * kD + d] = hin[row * kD + d] + acc;
}

// ---------------------------------------------------------------------------
// out[b, :] = hidden[tgt_idx[b], :]
// ---------------------------------------------------------------------------
__global__ __launch_bounds__(128)
void gather_out(const float* __restrict__ h, const int* __restrict__ tgt,
                float* __restrict__ out)
{
    const int b = blockIdx.x;
    const int d = threadIdx.x;
    out[b * kD + d] = h[tgt[b] * kD + d];
}

extern "C" void kernel_launch(void* const* d_in, const int* in_sizes, int n_in,
                              void* d_out, int out_size, void* d_ws, size_t ws_size,
                              hipStream_t stream)
{
    (void)in_sizes; (void)n_in; (void)out_size; (void)ws_size;

    const float* h0   = (const float*)d_in[0];   // hidden_state_list [1025,128]
    const int*   sp   = (const int*)  d_in[1];   // struct_parent [8,128,128]
    const int*   sc   = (const int*)  d_in[2];   // struct_child  [8,128,128]
    const int*   tgt  = (const int*)  d_in[3];   // tgt_idx [8]
    const float* Wi_f = (const float*)d_in[4];
    const float* Wh_f = (const float*)d_in[5];
    const float* bi_f = (const float*)d_in[6];
    const float* bh_f = (const float*)d_in[7];
    const float* Wi_b = (const float*)d_in[8];
    const float* Wh_b = (const float*)d_in[9];
    const float* bi_b = (const float*)d_in[10];
    const float* bh_b = (const float*)d_in[11];

    float* ws = (float*)d_ws;
    float* Gi = ws;                     // 1025*384
    float* Gh = Gi + kRows * kTD;       // 1025*384
    float* h1 = Gh + kRows * kTD;       // 1025*128
    float* h2 = h1 + kRows * kD;        // 1025*128   (total ~4.2 MB)

    const dim3 gemm_grid(kTilesM * kTilesN);

    // Phase 1: bwd GRU  (x = S_child, h = S_parent)
    gemm_hxt_wmma<<<gemm_grid, 32, 0, stream>>>(h0, Wi_b, bi_b, Gi);
    gemm_hxt_wmma<<<gemm_grid, 32, 0, stream>>>(h0, Wh_b, bh_b, Gh);
    gru_gather_phase<<<kRows, 128, 0, stream>>>(h0, Gi, Gh, sp, sc, h1, 1);

    // Phase 2: fwd GRU  (x = S_parent, h = S_child)
    gemm_hxt_wmma<<<gemm_grid, 32, 0, stream>>>(h1, Wi_f, bi_f, Gi);
    gemm_hxt_wmma<<<gemm_grid, 32, 0, stream>>>(h1, Wh_f, bh_f, Gh);
    gru_gather_phase<<<kRows, 128, 0, stream>>>(h1, Gi, Gh, sp, sc, h2, 0);

    // Final gather
    gather_out<<<kB, 128, 0, stream>>>(h2, tgt, (float*)d_out);
}